// SpodNet_86346022519495
// MI455X (gfx1250) — compile-verified
//
#include <hip/hip_runtime.h>
#include <math.h>

#define P 256
#define PB 65536        // P*P
#define NB 256          // batches

typedef float v2f __attribute__((ext_vector_type(2)));
typedef float v8f __attribute__((ext_vector_type(8)));

// D = A(16x4) * B(4x16) + C, fp32 WMMA (CDNA5, wave32)
__device__ __forceinline__ v8f wmma_f32_step(v8f c, v2f a, v2f b) {
    return __builtin_amdgcn_wmma_f32_16x16x4_f32(
        /*neg_a=*/false, a, /*neg_b=*/false, b,
        /*c_mod=*/(short)0, c, /*reuse_a=*/false, /*reuse_b=*/false);
}

// One b128 per matrix covers 8 logical K via K-permutation (A/B agree on the
// logical->hardware K mapping, so any bijection is valid). Works for LDS or
// global pointers (compiler selects ds_load_b128 vs global_load_b128).
__device__ __forceinline__ v8f wmma_k8(v8f acc, const float* arow,
                                       const float* brow, int base) {
    float4 av = *(const float4*)(arow + base);
    float4 bv = *(const float4*)(brow + base);
    v2f a0 = {av.x, av.y}, b0 = {bv.x, bv.y};
    v2f a1 = {av.z, av.w}, b1 = {bv.z, bv.w};
    acc = wmma_f32_step(acc, a0, b0);
    acc = wmma_f32_step(acc, a1, b1);
    return acc;
}

// ---------------- Prep: pad Wc (255x255) into zero-padded 256x256, once ----------------
__global__ void pad_wc(const float* __restrict__ Wc, float* __restrict__ WcPad) {
    int n = blockIdx.x, k = threadIdx.x;
    WcPad[n * P + k] = (n < P - 1 && k < P - 1) ? Wc[n * (P - 1) + k] : 0.f;
}

// ---------------- Pass 1 (fused gather + WMMA GEMM + symmetric scatter) ----------------
// Grid 16 x 512 threads: block owns one m-tile (16 batches); 16 waves cover all 16 n-tiles.
// Batch data is block-private, so gather->scatter ordering is handled by __syncthreads.
__global__ __launch_bounds__(512) void pass1_fused(const float* __restrict__ WcPad,
                                                   const float* __restrict__ bc,
                                                   float* __restrict__ Th, int col) {
    __shared__ float4 aLds4[16 * P / 4];            // 16KB: th12 rows for 16 batches
    float* aLds = (float*)aLds4;

    const int tid  = threadIdx.x;
    const int wave = tid >> 5;                      // n-tile 0..15
    const int lane = tid & 31;
    const int half = lane >> 4, l16 = lane & 15;
    const int m0 = blockIdx.x * 16;

    // Gather th12 (zero-padded at k=255) for this block's 16 batches.
    for (int e = tid; e < 16 * P; e += 512) {
        int mi = e >> 8, k = e & (P - 1);
        float val = 0.f;
        if (k < P - 1) {
            int j = k + (k >= col ? 1 : 0);
            val = Th[(m0 + mi) * PB + j * P + col];
        }
        aLds[mi * P + k] = val;
    }
    __syncthreads();

    const float* arow = aLds  + l16 * P + 4 * half;
    const float* brow = WcPad + (wave * 16 + l16) * P + 4 * half;
    v8f acc = {};
#pragma unroll 4
    for (int base = 0; base < P; base += 8)
        acc = wmma_k8(acc, arow, brow, base);

    const int n = wave * 16 + l16;
    if (n < P - 1) {
        const float bn = bc[n];
        const int jn = n + (n >= col ? 1 : 0);
#pragma unroll
        for (int r = 0; r < 8; ++r) {
            int m = m0 + r + 8 * half;               // batch index
            float val = 0.0625f * (acc[r] + bn);     // scale = 1/sqrt(256)
            Th[m * PB + jn * P + col] = val;
            Th[m * PB + col * P + jn] = val;
        }
    }
}

// ---------------- Pass 2 ----------------
// Stage-1 body (per batch, 256 threads): gather t,w12; s1=w12.t; m=W@t;
// v = m - w12*s1/w22; schur = t.v; feats = [th22, th12].
__device__ __forceinline__ void stage1_body(const float* __restrict__ Thb,
                                            const float* __restrict__ Wb,
                                            float* __restrict__ w12b,
                                            float* __restrict__ vb,
                                            float* __restrict__ featsb,
                                            float* __restrict__ scalb,
                                            int col, int j, float* t, float* red) {
    const float w22  = Wb[col * P + col];
    const float th22 = Thb[col * P + col];
    const float tj = (j == col) ? 0.f : Thb[j * P + col];
    const float wj = (j == col) ? 0.f : Wb[j * P + col];
    t[j] = tj;
    w12b[j] = wj;

    red[j] = wj * tj;                 // s1 = w12 . th12
    __syncthreads();
    for (int s = 128; s > 0; s >>= 1) {
        if (j < s) red[j] += red[j + s];
        __syncthreads();
    }
    const float s1 = red[0];
    __syncthreads();

    // m_j = row j of W times t (b128 row stream, L2-resident; W kept symmetric)
    float m = 0.f;
    const float4* rw = (const float4*)(Wb + j * P);
#pragma unroll 8
    for (int i = 0; i < P / 4; ++i) {
        float4 w4 = rw[i];
        m += w4.x * t[4 * i] + w4.y * t[4 * i + 1] + w4.z * t[4 * i + 2] + w4.w * t[4 * i + 3];
    }
    const float vj = (j == col) ? 0.f : (m - wj * (s1 / w22));
    vb[j] = vj;

    red[j] = tj * vj;                 // schur = th12 . v
    __syncthreads();
    for (int s = 128; s > 0; s >>= 1) {
        if (j < s) red[j] += red[j + s];
        __syncthreads();
    }
    if (j == 0) {
        scalb[0] = w22;
        scalb[2] = red[0];            // schur
    }

    // feats = [th22, th12]
    float f;
    if (j == 0) f = th22;
    else {
        int k = j - 1;
        f = t[k + (k >= col ? 1 : 0)];
    }
    featsb[j] = f;
}

// Standalone stage-1 (used once for col 0).
__global__ void pass2_stage1(const float* __restrict__ Th, const float* __restrict__ W,
                             float* __restrict__ w12f, float* __restrict__ vfull,
                             float* __restrict__ feats, float* __restrict__ scal, int col) {
    __shared__ float t[P];
    __shared__ float red[P];
    const int b = blockIdx.x, j = threadIdx.x;
    stage1_body(Th + b * PB, W + b * PB, w12f + b * P, vfull + b * P,
                feats + b * P, scal + b * 4, col, j, t, red);
}

// Fused 2-layer MLP + gy + diagonal write. Grid 16 x 512: block owns one m-tile
// (16 batches); 16 waves cover all 16 n-tiles; h1/h2 live only in LDS.
__global__ __launch_bounds__(512) void pass2_mlp_fused(
    const float* __restrict__ feats,
    const float* __restrict__ A1, const float* __restrict__ b1,
    const float* __restrict__ A2, const float* __restrict__ b2,
    const float* __restrict__ A3, const float* __restrict__ b3,
    float* __restrict__ scal, float* __restrict__ Th, int col) {
    __shared__ float4 aLds4[16 * P / 4];            // feats tile (16KB)
    __shared__ float4 h1Lds4[16 * P / 4];           // relu(layer1) (16KB)
    __shared__ float4 h2Lds4[16 * P / 4];           // relu(layer2) (16KB)
    float* aLds  = (float*)aLds4;
    float* h1Lds = (float*)h1Lds4;
    float* h2Lds = (float*)h2Lds4;

    const int tid  = threadIdx.x;
    const int wave = tid >> 5;                      // n-tile 0..15
    const int lane = tid & 31;
    const int half = lane >> 4, l16 = lane & 15;
    const int m0 = blockIdx.x * 16;
    const int n = wave * 16 + l16;

    // Stage feats rows (coalesced float4).
    for (int e = tid; e < 16 * P / 4; e += 512) {
        int mi = e >> 6, q = e & 63;
        aLds4[e] = ((const float4*)(feats + (m0 + mi) * P))[q];
    }
    __syncthreads();

    // Layer 1: h1 = relu(feats @ A1^T + b1)
    {
        const float* arow = aLds + l16 * P + 4 * half;
        const float* brow = A1 + n * P + 4 * half;
        v8f acc = {};
#pragma unroll 4
        for (int base = 0; base < P; base += 8)
            acc = wmma_k8(acc, arow, brow, base);
        const float bn = b1[n];
#pragma unroll
        for (int r = 0; r < 8; ++r) {
            float v = acc[r] + bn;
            h1Lds[(r + 8 * half) * P + n] = v > 0.f ? v : 0.f;
        }
    }
    __syncthreads();

    // Layer 2: h2 = relu(h1 @ A2^T + b2), A-fragments straight from LDS
    {
        const float* arow = h1Lds + l16 * P + 4 * half;
        const float* brow = A2 + n * P + 4 * half;
        v8f acc = {};
#pragma unroll 4
        for (int base = 0; base < P; base += 8)
            acc = wmma_k8(acc, arow, brow, base);
        const float bn = b2[n];
#pragma unroll
        for (int r = 0; r < 8; ++r) {
            float v = acc[r] + bn;
            h2Lds[(r + 8 * half) * P + n] = v > 0.f ? v : 0.f;
        }
    }
    __syncthreads();

    // gy[b] = exp(h2[b] . A3 + b3); wave w handles local batch row w.
    float partial = 0.f;
    const float* hr = h2Lds + wave * P;
#pragma unroll
    for (int k = lane; k < P; k += 32) partial += hr[k] * A3[k];
#pragma unroll
    for (int off = 16; off > 0; off >>= 1) partial += __shfl_xor(partial, off, 32);
    if (lane == 0) {
        const int b = m0 + wave;
        const float gy = expf(partial + b3[0]);
        scal[b * 4 + 3] = gy;
        Th[b * PB + col * P + col] = gy + scal[b * 4 + 2];
    }
}

// Fused per-batch: rank-2 W update for `col` (w12f/vfull are zero at index==col,
// so the rank-2 term leaves row/col `col` untouched) + row/col/diag overwrite,
// then — while W[b] is cache-hot — stage-1 for col+1.
__global__ void pass2_wupd_stage1(float* __restrict__ W, const float* __restrict__ Th,
                                  float* __restrict__ w12f, float* __restrict__ vfull,
                                  float* __restrict__ feats, float* __restrict__ scal,
                                  int col, int do_next) {
    __shared__ float ws_[P];
    __shared__ float vs_[P];
    __shared__ float t[P];
    __shared__ float red[P];
    const int b = blockIdx.x, tid = threadIdx.x;
    ws_[tid] = w12f[b * P + tid];
    vs_[tid] = vfull[b * P + tid];
    const float w22 = scal[b * 4 + 0];
    const float gy  = scal[b * 4 + 3];
    const float c1 = -1.0f / w22;
    const float c2 =  1.0f / gy;
    __syncthreads();

    float* Wb = W + b * PB;
    float4* W4 = (float4*)Wb;
    const float4* w4p = (const float4*)ws_;
    const float4* v4p = (const float4*)vs_;
    const int rsub = tid >> 6;        // 0..3: which of 4 rows this iteration
    const int c4   = tid & 63;        // float4 column index
    const float4 w4 = w4p[c4];
    const float4 v4 = v4p[c4];
    // Flat-coalesced: each iteration, the block touches 4 complete rows (4KB contiguous).
#pragma unroll 4
    for (int it = 0; it < P / 4; ++it) {
        const int row = it * 4 + rsub;
        const float a1 = c1 * ws_[row];
        const float a2 = c2 * vs_[row];
        float4* p = W4 + row * (P / 4) + c4;
        float4 r = *p;
        r.x += a1 * w4.x + a2 * v4.x;
        r.y += a1 * w4.y + a2 * v4.y;
        r.z += a1 * w4.z + a2 * v4.z;
        r.w += a1 * w4.w + a2 * v4.w;
        *p = r;
    }
    __syncthreads();
    if (tid != col) {
        const float w12n = -vs_[tid] * c2;     // -v/gy
        Wb[tid * P + col] = w12n;
        Wb[col * P + tid] = w12n;
    } else {
        Wb[col * P + col] = c2;                // 1/gy
    }

    if (do_next) {
        // Make this block's W[b] stores visible to all its threads, then stage-1
        // for the next column on the cache-hot matrix.
        __threadfence_block();
        __syncthreads();
        stage1_body(Th + b * PB, Wb, w12f + b * P, vfull + b * P,
                    feats + b * P, scal + b * 4, col + 1, tid, t, red);
    }
}

extern "C" void kernel_launch(void* const* d_in, const int* in_sizes, int n_in,
                              void* d_out, int out_size, void* d_ws, size_t ws_size,
                              hipStream_t stream) {
    const float* Theta  = (const float*)d_in[0];
    const float* W_init = (const float*)d_in[1];
    const float* Wc = (const float*)d_in[2];
    const float* bc = (const float*)d_in[3];
    const float* A1 = (const float*)d_in[4];
    const float* b1 = (const float*)d_in[5];
    const float* A2 = (const float*)d_in[6];
    const float* b2 = (const float*)d_in[7];
    const float* A3 = (const float*)d_in[8];
    const float* b3 = (const float*)d_in[9];

    float* Th = (float*)d_out;
    float* ws = (float*)d_ws;
    float* Wmat  = ws;                         // 256*256*256
    float* w12f  = Wmat  + (size_t)NB * PB;    // 256*256
    float* vfull = w12f  + NB * P;
    float* feats = vfull + NB * P;
    float* scal  = feats + NB * P;             // 256*4
    float* WcPad = scal  + NB * 4;             // 256*256 zero-padded Wc

    hipMemcpyAsync(Th,   Theta,  (size_t)NB * PB * sizeof(float), hipMemcpyDeviceToDevice, stream);
    hipMemcpyAsync(Wmat, W_init, (size_t)NB * PB * sizeof(float), hipMemcpyDeviceToDevice, stream);
    pad_wc<<<P, P, 0, stream>>>(Wc, WcPad);

    for (int col = 0; col < P; ++col)
        pass1_fused<<<16, 512, 0, stream>>>(WcPad, bc, Th, col);

    pass2_stage1<<<NB, P, 0, stream>>>(Th, Wmat, w12f, vfull, feats, scal, 0);
    for (int col = 0; col < P; ++col) {
        pass2_mlp_fused<<<16, 512, 0, stream>>>(feats, A1, b1, A2, b2, A3, b3, scal, Th, col);
        pass2_wupd_stage1<<<NB, P, 0, stream>>>(Wmat, Th, w12f, vfull, feats, scal,
                                                col, col < P - 1 ? 1 : 0);
    }
}